// GaussianEmbedding_12249246728257
// MI455X (gfx1250) — compile-verified
//
#include <hip/hip_runtime.h>
#include <hip/hip_bf16.h>

typedef __attribute__((ext_vector_type(2))) float v2f;
typedef __attribute__((ext_vector_type(4))) float v4f;
typedef __attribute__((ext_vector_type(8))) float v8f;

// ---------------------------------------------------------------------------
// C[b] = A[b] @ B[b], N x N fp32, N multiple of 128.
// Block tile 128(M) x 128(N), K-step 16, 256 threads = 8 waves (wave32),
// waves 4(M) x 2(N): each wave owns 32(M) x 64(N) -> 8 accumulators.
// Double-buffered LDS pipeline:
//   - A tile:  GLOBAL_LOAD_ASYNC_TO_LDS_B128 (ASYNCcnt DMA, no VGPR staging)
//   - B tile:  register-staged + pair-interleaved LDS scatter (keeps WMMA
//              B-operand loads as single aligned ds_load_b64)
//   - one s_wait_asynccnt + barrier per K-step
//
// V_WMMA_F32_16X16X4_F32 operand layouts (CDNA5 ISA 7.12.2):
//   A 16x4 : M = lane&15 ; VGPR v holds K = v + 2*(lane>>4)   (pair contig)
//   B 4x16 : N = lane&15 ; VGPR v holds K = v + 2*(lane>>4)   (pair contig
//            in the pair-interleaved LDS layout Bs[k/2][n][2])
//   C 16x16: N = lane&15 ; VGPR v holds M = v + 8*(lane>>4)
// ---------------------------------------------------------------------------
typedef float AsTile[128][20];      // 128 x 16 A tile, rows padded to 80B:
                                    // 16B-aligned rows, 16 distinct banks
typedef float BsTile[8][144][2];    // k-pair-interleaved 16 x 128 B tile;
                                    // row stride 288 floats -> hi half-wave
                                    // lands on banks +32 (conflict-free b64)

__device__ __forceinline__ void async_copy_b128(unsigned ldsAddr,
                                                const float* gptr) {
    asm volatile("global_load_async_to_lds_b128 %0, %1, off"
                 :
                 : "v"(ldsAddr), "v"((unsigned long long)(uintptr_t)gptr)
                 : "memory");
}

__device__ __forceinline__ void wait_async_all() {
    asm volatile("s_wait_asynccnt 0x0" ::: "memory");
}

__global__ __launch_bounds__(256)
void power_matmul_f32(const float* __restrict__ Abase,
                      const float* __restrict__ Bbase,
                      float* __restrict__ Cbase,
                      int N)
{
    const size_t matElems = (size_t)N * (size_t)N;
    const float* Am = Abase + (size_t)blockIdx.z * matElems;
    const float* Bm = Bbase + (size_t)blockIdx.z * matElems;
    float*       Cm = Cbase + (size_t)blockIdx.z * matElems;

    __shared__ __align__(16) AsTile AsBuf[2];
    __shared__ __align__(16) BsTile BsBuf[2];

    const int t    = threadIdx.x;        // 0..255
    const int lane = t & 31;
    const int wave = t >> 5;             // 0..7
    const int hi   = lane >> 4;          // 0/1
    const int lo   = lane & 15;
    const int wm   = wave >> 1;          // 0..3 : M strip (32 rows)
    const int wn   = wave & 1;           // 0..1 : N half (64 cols)

    const int rowA0 = blockIdx.y * 128;
    const int colB0 = blockIdx.x * 128;

    // Per-thread tile-load coordinates (512 float4 slots each, 2 per thread).
    const int arA0 = t >> 2,         acA0 = (t & 3) * 4;          // A slot t
    const int arA1 = (t + 256) >> 2, acA1 = ((t + 256) & 3) * 4;  // A slot t+256
    const int brB0 = t >> 5,         bcB0 = (t & 31) * 4;         // B slot t
    const int brB1 = (t + 256) >> 5, bcB1 = ((t + 256) & 31) * 4; // B slot t+256

    // LDS destinations for the two async A copies, per buffer.
    const unsigned aDst0[2] = { (unsigned)(uintptr_t)&AsBuf[0][arA0][acA0],
                                (unsigned)(uintptr_t)&AsBuf[1][arA0][acA0] };
    const unsigned aDst1[2] = { (unsigned)(uintptr_t)&AsBuf[0][arA1][acA1],
                                (unsigned)(uintptr_t)&AsBuf[1][arA1][acA1] };

    v8f acc[2][4];
#pragma unroll
    for (int mi = 0; mi < 2; ++mi)
#pragma unroll
        for (int j = 0; j < 4; ++j) acc[mi][j] = (v8f)0.0f;

    // ---- preload first K-tile into buffer 0 ----
    {
        async_copy_b128(aDst0[0], Am + (size_t)(rowA0 + arA0) * N + acA0);
        async_copy_b128(aDst1[0], Am + (size_t)(rowA0 + arA1) * N + acA1);
        v4f b0 = *(const v4f*)(Bm + (size_t)brB0 * N + colB0 + bcB0);
        v4f b1 = *(const v4f*)(Bm + (size_t)brB1 * N + colB0 + bcB1);
#pragma unroll
        for (int i = 0; i < 4; ++i) {
            BsBuf[0][brB0 >> 1][bcB0 + i][brB0 & 1] = b0[i];
            BsBuf[0][brB1 >> 1][bcB1 + i][brB1 & 1] = b1[i];
        }
    }
    wait_async_all();
    __syncthreads();

    int p = 0;
    for (int kb = 0; kb < N; kb += 16) {
        const bool more = (kb + 16) < N;
        const int  q    = p ^ 1;

        // ---- kick off next tile: async DMA for A, register-stage B ----
        v4f b0, b1;
        if (more) {
            const int kn = kb + 16;
            async_copy_b128(aDst0[q], Am + (size_t)(rowA0 + arA0) * N + kn + acA0);
            async_copy_b128(aDst1[q], Am + (size_t)(rowA0 + arA1) * N + kn + acA1);
            b0 = *(const v4f*)(Bm + (size_t)(kn + brB0) * N + colB0 + bcB0);
            b1 = *(const v4f*)(Bm + (size_t)(kn + brB1) * N + colB0 + bcB1);
        }

        // ---- 4 K-slices of 4 : 8 WMMAs each = 32 per wave per K-step ----
        const AsTile& As = AsBuf[p];
        const BsTile& Bs = BsBuf[p];
#pragma unroll
        for (int ks = 0; ks < 4; ++ks) {
            const int k0 = ks * 4 + 2 * hi;
            v2f a_0 = *(const v2f*)&As[wm * 32 +      lo][k0];
            v2f a_1 = *(const v2f*)&As[wm * 32 + 16 + lo][k0];
#pragma unroll
            for (int j = 0; j < 4; ++j) {
                v2f b = *(const v2f*)&Bs[ks * 2 + hi][wn * 64 + j * 16 + lo][0];
                acc[0][j] = __builtin_amdgcn_wmma_f32_16x16x4_f32(
                    false, a_0, false, b, (short)0, acc[0][j], false, false);
                acc[1][j] = __builtin_amdgcn_wmma_f32_16x16x4_f32(
                    false, a_1, false, b, (short)0, acc[1][j], false, false);
            }
        }

        // ---- commit staged B tile to the alternate LDS buffer ----
        if (more) {
#pragma unroll
            for (int i = 0; i < 4; ++i) {
                BsBuf[q][brB0 >> 1][bcB0 + i][brB0 & 1] = b0[i];
                BsBuf[q][brB1 >> 1][bcB1 + i][brB1 & 1] = b1[i];
            }
        }
        wait_async_all();
        __syncthreads();
        p = q;
    }

    // ---- store C: lane covers N = lo, VGPR v covers M = v + 8*hi ----
#pragma unroll
    for (int mi = 0; mi < 2; ++mi) {
        const int mBase = rowA0 + wm * 32 + mi * 16 + 8 * hi;
#pragma unroll
        for (int j = 0; j < 4; ++j) {
            const int nIdx = colB0 + wn * 64 + j * 16 + lo;
#pragma unroll
            for (int v = 0; v < 8; ++v) {
                Cm[(size_t)(mBase + v) * N + nIdx] = acc[mi][j][v];
            }
        }
    }
}

// ---------------------------------------------------------------------------
// Per (b, n): gather d0..d8 and emit Y[b,n,c] = sum_t h[t,c] * d_t.
//   d0 = 1, d1..d4 = diagonals of A,P2,P3,P4
//   d5 = P2[n,:].P3[:,n], d6 = P3.P3, d7 = P3.P4, d8 = P4.P4
// ---------------------------------------------------------------------------
__global__ __launch_bounds__(256)
void diag_combine(const float* __restrict__ Ab,
                  const float* __restrict__ P2,
                  const float* __restrict__ P3,
                  const float* __restrict__ P4,
                  const float* __restrict__ h,    // [9, C]
                  float* __restrict__ Y,          // [B, N, C]
                  int b0, int N, int C)
{
    const int n  = blockIdx.x;
    const int lb = blockIdx.y;
    const int b  = b0 + lb;
    const size_t mo = (size_t)lb * (size_t)N * (size_t)N;
    const float* a2 = P2 + mo;
    const float* a3 = P3 + mo;
    const float* a4 = P4 + mo;

    const int t = threadIdx.x;
    float s5 = 0.f, s6 = 0.f, s7 = 0.f, s8 = 0.f;
    for (int m = t; m < N; m += 256) {
        const float r2 = a2[(size_t)n * N + m];
        const float r3 = a3[(size_t)n * N + m];
        const float r4 = a4[(size_t)n * N + m];
        const float c3 = a3[(size_t)m * N + n];
        const float c4 = a4[(size_t)m * N + n];
        s5 += r2 * c3;
        s6 += r3 * c3;
        s7 += r3 * c4;
        s8 += r4 * c4;
    }

    __shared__ float red[4][256];
    red[0][t] = s5; red[1][t] = s6; red[2][t] = s7; red[3][t] = s8;
    __syncthreads();
    for (int off = 128; off > 0; off >>= 1) {
        if (t < off) {
#pragma unroll
            for (int q = 0; q < 4; ++q) red[q][t] += red[q][t + off];
        }
        __syncthreads();
    }

    __shared__ float dvals[9];
    if (t == 0) {
        dvals[0] = 1.0f;
        dvals[1] = Ab[mo + (size_t)n * N + n];
        dvals[2] = a2[(size_t)n * N + n];
        dvals[3] = a3[(size_t)n * N + n];
        dvals[4] = a4[(size_t)n * N + n];
        dvals[5] = red[0][0];
        dvals[6] = red[1][0];
        dvals[7] = red[2][0];
        dvals[8] = red[3][0];
    }
    __syncthreads();

    if (t < C) {
        float y = 0.f;
#pragma unroll
        for (int q = 0; q < 9; ++q) y += h[q * C + t] * dvals[q];
        Y[((size_t)b * N + n) * C + t] = y;
    }
}

// ---------------------------------------------------------------------------
// Orchestration: per batch-chunk (fits in d_ws):
//   P2 = A@A ; P3 = P2@A ; P4 = P2@P2 ; then diag_combine -> Y
// ---------------------------------------------------------------------------
extern "C" void kernel_launch(void* const* d_in, const int* in_sizes, int n_in,
                              void* d_out, int out_size, void* d_ws, size_t ws_size,
                              hipStream_t stream) {
    const int N = 1024;
    const int C = 32;
    const float* A = (const float*)d_in[0];
    const float* h = (const float*)d_in[1];
    float* Y = (float*)d_out;

    const int B = in_sizes[0] / (N * N);
    const size_t matElems = (size_t)N * (size_t)N;
    const size_t matBytes = matElems * sizeof(float);

    int maxNb = (int)(ws_size / (3 * matBytes));
    if (maxNb < 1) maxNb = 1;
    if (maxNb > B) maxNb = B;

    float* P2 = (float*)d_ws;
    float* P3 = P2 + (size_t)maxNb * matElems;
    float* P4 = P3 + (size_t)maxNb * matElems;

    for (int b0 = 0; b0 < B; b0 += maxNb) {
        const int nb = (B - b0 < maxNb) ? (B - b0) : maxNb;
        const float* Ab = A + (size_t)b0 * matElems;

        dim3 blk(256);
        dim3 grd(N / 128, N / 128, nb);
        power_matmul_f32<<<grd, blk, 0, stream>>>(Ab, Ab, P2, N); // A^2
        power_matmul_f32<<<grd, blk, 0, stream>>>(P2, Ab, P3, N); // A^3
        power_matmul_f32<<<grd, blk, 0, stream>>>(P2, P2, P4, N); // A^4

        dim3 grd2(N, nb);
        diag_combine<<<grd2, 256, 0, stream>>>(Ab, P2, P3, P4, h, Y, b0, N, C);
    }
}